// EMDLoss_59914793779671
// MI455X (gfx1250) — compile-verified
//
#include <hip/hip_runtime.h>
#include <math.h>

// Problem constants (match reference)
#define N        128
#define NP       129      // with extra "energy difference" bin
#define NT       144      // padded to 9 tiles of 16
#define KSTRIDE  145      // LDS row stride in floats (odd -> bank-staggered)
#define INV_EPS  50.0f    // 1/0.02
#define ITERS    100
#define NWAVES   9
#define NTHREADS (NWAVES * 32)

typedef __attribute__((ext_vector_type(2))) float v2f;
typedef __attribute__((ext_vector_type(8))) float v8f;

__launch_bounds__(NTHREADS)
__global__ void emd_sinkhorn_wmma(const float* __restrict__ jets1,
                                  const float* __restrict__ jets2,
                                  float* __restrict__ out)
{
    // K matrix in f32: exp(-C/eps) spans e^0 .. ~2e-31 -> needs f32 range.
    __shared__ float Ksh[NT * KSTRIDE];             // ~84 KB
    __shared__ float x1s[N], y1s[N], w1s[N];
    __shared__ float x2s[N], y2s[N], w2s[N];
    __shared__ float aa[NT], bb[NT];                // marginals (padded with 0)
    __shared__ float ua[NT], va[NT];                // scaling vectors u, v
    __shared__ float red[NWAVES];
    __shared__ float EabsSh;

    const int b    = blockIdx.x;
    const int tid  = threadIdx.x;
    const int lane = tid & 31;
    const int wave = tid >> 5;      // 0..8 : one 16-wide tile per wave
    const int half = lane >> 4;     // 0 for lanes 0-15, 1 for lanes 16-31
    const int l15  = lane & 15;

    // ---- stage jets into LDS -------------------------------------------------
    const float* j1 = jets1 + (size_t)b * N * 3;
    const float* j2 = jets2 + (size_t)b * N * 3;
    for (int i = tid; i < N; i += NTHREADS) {
        x1s[i] = j1[3 * i + 0];  y1s[i] = j1[3 * i + 1];  w1s[i] = j1[3 * i + 2];
        x2s[i] = j2[3 * i + 0];  y2s[i] = j2[3 * i + 1];  w2s[i] = j2[3 * i + 2];
    }
    __syncthreads();

    // ---- marginals a, b and |E2-E1| -----------------------------------------
    if (tid == 0) {
        float E1 = 0.f, E2 = 0.f;
        for (int i = 0; i < N; ++i) { E1 += w1s[i]; E2 += w2s[i]; }
        EabsSh = fabsf(E2 - E1);
        aa[N]  = fmaxf(E2 - E1, 0.f);
        bb[N]  = fmaxf(E1 - E2, 0.f);
    }
    for (int i = tid; i < NT; i += NTHREADS) {
        if (i < N)      { aa[i] = w1s[i]; bb[i] = w2s[i]; }
        else if (i > N) { aa[i] = 0.f;    bb[i] = 0.f;    }
        ua[i] = 0.f;
        va[i] = (i < NP) ? 1.f : 0.f;   // g=0  =>  v=exp(g/eps)=1 on real bins
    }
    __syncthreads();

    // ---- K = exp(-C/eps); Cp pad row/col (index 128) has C=0 -> K=1;
    //      synthetic pad (129..143) gets K=0 so it is inert. -------------------
    for (int idx = tid; idx < NT * NT; idx += NTHREADS) {
        const int i = idx / NT;
        const int j = idx - i * NT;
        float k;
        if (i < NP && j < NP) {
            if (i < N && j < N) {
                const float dx = -(x1s[i] - x2s[j]) + 1e-12f;
                const float dy = -(y1s[i] - y2s[j]) + 1e-12f;
                const float d  = sqrtf(dx * dx + dy * dy);
                k = __expf(-d * INV_EPS);
            } else {
                k = 1.0f;
            }
        } else {
            k = 0.0f;
        }
        Ksh[i * KSTRIDE + j] = k;
    }
    __syncthreads();

    // ---- Sinkhorn in scaling form, matvecs via V_WMMA_F32_16X16X4_F32 --------
    // A 16x4 f32 layout: lane l holds A[l&15][k0 + 2*(l>=16) + {0,1}] in 2 VGPRs.
    // B 4x16  f32 layout: VGPR0 = rows {0,2}, VGPR1 = rows {1,3} across lane halves.
    for (int it = 0; it < ITERS; ++it) {
        // r = K * v  ->  u = a / r        (B = v broadcast across all 16 columns)
        {
            v8f acc = {};
            const int ti = wave;
            for (int k0 = 0; k0 < NT; k0 += 4) {
                v2f A, B;
                const float* arow = &Ksh[(ti * 16 + l15) * KSTRIDE + k0 + 2 * half];
                A.x = arow[0];
                A.y = arow[1];
                B.x = va[k0 + 2 * half + 0];
                B.y = va[k0 + 2 * half + 1];
                acc = __builtin_amdgcn_wmma_f32_16x16x4_f32(
                    false, A, false, B, (short)0, acc, false, false);
            }
            // D[m][n] identical over n; lane 0 holds rows ti*16+0..7, lane 16 rows +8..15
            if (l15 == 0) {
                const int base = ti * 16 + 8 * half;
                #pragma unroll
                for (int r = 0; r < 8; ++r) {
                    const float rs = acc[r];
                    const float av = aa[base + r];
                    ua[base + r] = (av > 0.f && rs > 0.f) ? (av / rs) : 0.f;
                }
            }
        }
        __syncthreads();

        // s = K^T * u  ->  v = b / s      (A = u broadcast across all 16 rows)
        {
            v8f acc = {};
            const int tj = wave;
            for (int k0 = 0; k0 < NT; k0 += 4) {
                v2f A, B;
                A.x = ua[k0 + 2 * half + 0];
                A.y = ua[k0 + 2 * half + 1];
                const float* bcol = &Ksh[(k0 + 2 * half) * KSTRIDE + tj * 16 + l15];
                B.x = bcol[0];
                B.y = bcol[KSTRIDE];
                acc = __builtin_amdgcn_wmma_f32_16x16x4_f32(
                    false, A, false, B, (short)0, acc, false, false);
            }
            // D[m][n] identical over m; row 0, lanes 0..15 cover columns tj*16+n
            if (lane < 16) {
                const float cs = acc[0];
                const float bv = bb[tj * 16 + lane];
                va[tj * 16 + lane] = (bv > 0.f && cs > 0.f) ? (bv / cs) : 0.f;
            }
        }
        __syncthreads();
    }

    // ---- emd = sum_{i,j<128} dist(i,j) * u_i * K_ij * v_j  + |E2-E1| ---------
    float local = 0.f;
    for (int idx = tid; idx < N * N; idx += NTHREADS) {
        const int i = idx >> 7;
        const int j = idx & (N - 1);
        const float dx = -(x1s[i] - x2s[j]) + 1e-12f;
        const float dy = -(y1s[i] - y2s[j]) + 1e-12f;
        const float d  = sqrtf(dx * dx + dy * dy);
        local += d * ua[i] * Ksh[i * KSTRIDE + j] * va[j];
    }
    #pragma unroll
    for (int off = 16; off > 0; off >>= 1)
        local += __shfl_down(local, off, 32);
    if (lane == 0) red[wave] = local;
    __syncthreads();
    if (tid == 0) {
        float s = 0.f;
        for (int w = 0; w < NWAVES; ++w) s += red[w];
        out[b] = s + EabsSh;
    }
}

extern "C" void kernel_launch(void* const* d_in, const int* in_sizes, int n_in,
                              void* d_out, int out_size, void* d_ws, size_t ws_size,
                              hipStream_t stream) {
    const float* jets1 = (const float*)d_in[0];
    const float* jets2 = (const float*)d_in[1];
    float* out = (float*)d_out;
    const int batch = in_sizes[0] / (N * 3);   // 1024
    emd_sinkhorn_wmma<<<dim3(batch), dim3(NTHREADS), 0, stream>>>(jets1, jets2, out);
}